// CaputoFractionalActivation_58677843198095
// MI455X (gfx1250) — compile-verified
//
#include <hip/hip_runtime.h>
#include <math.h>

#define MIN_STEP 1e-6f
#define MAX_STEP 1e-3f
#define RED_BLOCK 256

// ---------------- gfx1250 feature detection ----------------
#if __has_builtin(__builtin_amdgcn_global_load_async_to_lds_b128)
#define HAVE_ASYNC_LDS 1
#else
#define HAVE_ASYNC_LDS 0
#endif

// ---------------- fast device helpers ----------------

__device__ __forceinline__ float tanh_fast(float x) {
#if __has_builtin(__builtin_amdgcn_tanhf)
    return __builtin_amdgcn_tanhf(x);   // gfx1250 hardware v_tanh_f32
#else
    return tanhf(x);
#endif
}

__device__ __forceinline__ float rcp_fast(float x) {
#if __has_builtin(__builtin_amdgcn_rcpf)
    return __builtin_amdgcn_rcpf(x);    // v_rcp_f32
#else
    return 1.0f / x;
#endif
}

// Monotone float -> uint key so we can use integer atomicMin/atomicMax.
__device__ __forceinline__ unsigned orderKey(float f) {
    unsigned u = __float_as_uint(f);
    return (u & 0x80000000u) ? ~u : (u | 0x80000000u);
}
__device__ __forceinline__ float unorderKey(unsigned k) {
    return (k & 0x80000000u) ? __uint_as_float(k & 0x7FFFFFFFu)
                             : __uint_as_float(~k);
}

#if HAVE_ASYNC_LDS
typedef int v4i __attribute__((ext_vector_type(4)));
typedef __attribute__((address_space(1))) v4i as1_v4i;   // global
typedef __attribute__((address_space(3))) v4i as3_v4i;   // LDS

__device__ __forceinline__ void async_copy16(const float4* gsrc, float4* ldst) {
    // GLOBAL_LOAD_ASYNC_TO_LDS_B128: memory -> LDS, tracked by ASYNCcnt
    __builtin_amdgcn_global_load_async_to_lds_b128(
        (as1_v4i*)gsrc, (as3_v4i*)ldst, /*offset=*/0, /*cpol=*/0);
}

template <int N>
__device__ __forceinline__ void wait_asynccnt() {
#if __has_builtin(__builtin_amdgcn_s_wait_asynccnt)
    __builtin_amdgcn_s_wait_asynccnt(N);
#else
    asm volatile("s_wait_asynccnt %0" ::"i"(N) : "memory");
#endif
}

__device__ __forceinline__ void wait_dscnt0() {
    asm volatile("s_wait_dscnt 0x0" ::: "memory");
}
#endif

// ---------- kernel 2 (defined first so it leads the asm dump) ----------
// Caputo series: c0 = 1; c_{k+1} = -c_k * (alpha + k + 1) / (k + 1), alpha = 0.5
__device__ __forceinline__ float caputo_one(float v, float xmin, float hscale) {
    const float C[10] = {
         1.0f,               -1.5f,
         1.875f,             -2.1875f,
         2.4609375f,         -2.70703125f,
         2.9326171875f,      -3.14208984375f,
         3.338470458984375f, -3.5239410400390625f
    };
    const float h    = fmaf(v - xmin, hscale, MIN_STEP);
    const float base = tanh_fast(v);

    float prev = base;
    float acc  = 0.0f;
    float xk   = v;
    #pragma unroll
    for (int k = 0; k < 10; ++k) {
        xk  -= h;
        const float cur = tanh_fast(xk);      // v_tanh_f32
        acc  = fmaf(C[k], prev - cur, acc);
        prev = cur;
    }
    return base + acc * rcp_fast(h);
}

__global__ void caputo_main(const float* __restrict__ x,
                            float* __restrict__ out, int n,
                            const unsigned* __restrict__ ws) {
    const float xmax   = unorderKey(ws[0]);
    const float xmin   = unorderKey(ws[1]);
    // v_rcp_f32-based scale (exact divide is overkill: h itself uses rcp)
    const float hscale = (MAX_STEP - MIN_STEP) * rcp_fast(xmax - xmin);

    const int n4  = n >> 2;
    const int tid = blockIdx.x * blockDim.x + threadIdx.x;

    if (tid < n4) {
        float4 v = ((const float4*)x)[tid];               // global_load_b128
        float4 r;
        r.x = caputo_one(v.x, xmin, hscale);
        r.y = caputo_one(v.y, xmin, hscale);
        r.z = caputo_one(v.z, xmin, hscale);
        r.w = caputo_one(v.w, xmin, hscale);
        ((float4*)out)[tid] = r;                          // global_store_b128
    } else {
        const int i = (n4 << 2) + (tid - n4);             // tail (n % 4)
        if (i < n) out[i] = caputo_one(x[i], xmin, hscale);
    }
}

// ---------- kernel 1: global min/max reduction ----------
// Streams x through LDS with double-buffered async copies (ASYNCcnt pipeline).
__global__ void caputo_minmax(const float* __restrict__ x, int n,
                              unsigned* __restrict__ ws) {
    const int tid    = blockIdx.x * blockDim.x + threadIdx.x;
    const int stride = gridDim.x * blockDim.x;
    const int n4     = n >> 2;

    float vmin =  INFINITY;
    float vmax = -INFINITY;

    const float4* __restrict__ x4 = (const float4*)x;

#if HAVE_ASYNC_LDS
    __shared__ float4 stage[2][RED_BLOCK];   // 8 KB double buffer
    int i = tid;
    int p = 0;
    if (i < n4) async_copy16(&x4[i], &stage[0][threadIdx.x]);
    while (i < n4) {
        const int  nxt      = i + stride;
        const bool havenext = nxt < n4;
        if (havenext) async_copy16(&x4[nxt], &stage[p ^ 1][threadIdx.x]);
        if (havenext) wait_asynccnt<1>();    // current buffer landed (in-order)
        else          wait_asynccnt<0>();
        const float4 v = stage[p][threadIdx.x];          // ds_load_b128
        vmin = fminf(vmin, fminf(fminf(v.x, v.y), fminf(v.z, v.w)));
        vmax = fmaxf(vmax, fmaxf(fmaxf(v.x, v.y), fmaxf(v.z, v.w)));
        wait_dscnt0();   // ds_read done before async engine may recycle buffer p
        p ^= 1;
        i = nxt;
    }
#else
    for (int i = tid; i < n4; i += stride) {
        float4 v = x4[i];
        vmin = fminf(vmin, fminf(fminf(v.x, v.y), fminf(v.z, v.w)));
        vmax = fmaxf(vmax, fmaxf(fmaxf(v.x, v.y), fmaxf(v.z, v.w)));
    }
#endif

    for (int i = (n4 << 2) + tid; i < n; i += stride) {
        float v = x[i];
        vmin = fminf(vmin, v);
        vmax = fmaxf(vmax, v);
    }

    // wave32 butterfly reduction
    #pragma unroll
    for (int off = 16; off > 0; off >>= 1) {
        vmin = fminf(vmin, __shfl_xor(vmin, off, 32));
        vmax = fmaxf(vmax, __shfl_xor(vmax, off, 32));
    }

    __shared__ float smin[8], smax[8];   // 256 threads = 8 wave32
    const int lane = threadIdx.x & 31;
    const int wave = threadIdx.x >> 5;
    if (lane == 0) { smin[wave] = vmin; smax[wave] = vmax; }
    __syncthreads();

    if (wave == 0) {
        float m0 = (lane < 8) ? smin[lane] :  INFINITY;
        float m1 = (lane < 8) ? smax[lane] : -INFINITY;
        #pragma unroll
        for (int off = 4; off > 0; off >>= 1) {
            m0 = fminf(m0, __shfl_xor(m0, off, 32));
            m1 = fmaxf(m1, __shfl_xor(m1, off, 32));
        }
        if (lane == 0) {
            atomicMax(&ws[0], orderKey(m1));
            atomicMin(&ws[1], orderKey(m0));
        }
    }
}

// ---------- kernel 0: init workspace (stateless per-call) ----------
__global__ void caputo_init_ws(unsigned* ws) {
    ws[0] = 0u;           // max-key slot (identity for unsigned max)
    ws[1] = 0xFFFFFFFFu;  // min-key slot
}

// ---------------- launch ----------------
extern "C" void kernel_launch(void* const* d_in, const int* in_sizes, int n_in,
                              void* d_out, int out_size, void* d_ws, size_t ws_size,
                              hipStream_t stream) {
    const float* x   = (const float*)d_in[0];
    float*       out = (float*)d_out;
    unsigned*    ws  = (unsigned*)d_ws;
    const int n = in_sizes[0];

    caputo_init_ws<<<1, 1, 0, stream>>>(ws);

    const int rblocks = 2048;   // grid-stride streaming reduction (8 iters/thread)
    caputo_minmax<<<rblocks, RED_BLOCK, 0, stream>>>(x, n, ws);

    const int n4     = n >> 2;
    const int tail   = n - (n4 << 2);
    const int items  = n4 + tail;
    const int mblocks = (items + 255) / 256;
    caputo_main<<<mblocks, 256, 0, stream>>>(x, out, n, ws);
}